// MyNewGCN_25890062860848
// MI455X (gfx1250) — compile-verified
//
#include <hip/hip_runtime.h>
#include <hip/hip_bf16.h>

// ---------------------------------------------------------------------------
// GCN forward for MI455X (gfx1250, wave32). All GEMMs via v_wmma_f32_16x16x4.
// ---------------------------------------------------------------------------

#define NNODES 100000
#define NGRAPH 512

typedef __attribute__((ext_vector_type(2))) float v2f;
typedef __attribute__((ext_vector_type(8))) float v8f;

__device__ __forceinline__ v8f wmma4(v2f a, v2f b, v8f c) {
  // D = A(16x4 f32) * B(4x16 f32) + C(16x16 f32)
  return __builtin_amdgcn_wmma_f32_16x16x4_f32(
      /*neg_a=*/false, a, /*neg_b=*/false, b,
      /*c_mod=*/(short)0, c, /*reuse_a=*/false, /*reuse_b=*/false);
}

__device__ __forceinline__ void atomAddF(float* p, float v) {
  // lowers to global_atomic_add_f32 (relaxed, device scope)
  __hip_atomic_fetch_add(p, v, __ATOMIC_RELAXED, __HIP_MEMORY_SCOPE_AGENT);
}

// ---------------- degree / norm -------------------------------------------
__global__ void deg_kernel(const int* __restrict__ idx, float* __restrict__ deg,
                           int n_edges) {
  int i = blockIdx.x * blockDim.x + threadIdx.x;
  if (i < n_edges) atomAddF(&deg[idx[i]], 1.0f);
}

__global__ void rsqrt_kernel(float* __restrict__ d, int n) {
  int i = blockIdx.x * blockDim.x + threadIdx.x;
  if (i < n) d[i] = rsqrtf(fmaxf(d[i], 1.0f));
}

// ---------------- node GEMM: C[M,N] = (A ⊙ s) @ W --------------------------
// one wave per 16-row m-tile, full K reduction in-wave, N/16 n-tiles.
template <int K, int N>
__global__ __launch_bounds__(32) void gemm_node_scaled(
    const float* __restrict__ A, const float* __restrict__ s,
    const float* __restrict__ W, float* __restrict__ C) {
  const int lane = threadIdx.x;
  const int half = lane >> 4;   // 0: K pair {0,1}, 1: K pair {2,3}
  const int lm   = lane & 15;   // A row / B-D column within tile
  const int m0   = blockIdx.x * 16;
  constexpr int NT = N / 16;

  const float sc = s[m0 + lm];
  const float* Arow = A + (size_t)(m0 + lm) * K;

  v8f acc[NT] = {};
#pragma unroll 4
  for (int k0 = 0; k0 < K; k0 += 4) {
    const int ka = k0 + 2 * half;
    v2f a;
    a.x = Arow[ka + 0] * sc;
    a.y = Arow[ka + 1] * sc;
#pragma unroll
    for (int t = 0; t < NT; ++t) {
      v2f b;
      b.x = W[(size_t)(ka + 0) * N + t * 16 + lm];
      b.y = W[(size_t)(ka + 1) * N + t * 16 + lm];
      acc[t] = wmma4(a, b, acc[t]);
    }
  }
#pragma unroll
  for (int t = 0; t < NT; ++t)
#pragma unroll
    for (int r = 0; r < 8; ++r)
      C[(size_t)(m0 + r + 8 * half) * N + t * 16 + lm] = acc[t][r];
}

// ---------------- edge scatter-add ----------------------------------------
__global__ void scatter_kernel(const float* __restrict__ h,
                               const int* __restrict__ src,
                               const int* __restrict__ dst,
                               float* __restrict__ agg,
                               int n_edges, int lognf) {
  long long t = (long long)blockIdx.x * blockDim.x + threadIdx.x;
  if (t >= ((long long)n_edges << lognf)) return;
  int e = (int)(t >> lognf);
  int f = (int)(t & ((1 << lognf) - 1));
  atomAddF(&agg[((long long)dst[e] << lognf) + f],
           h[((long long)src[e] << lognf) + f]);
}

// ---------------- gconv epilogue: out = act(agg ⊙ dn + b) ------------------
__global__ void epilogue_kernel(const float* __restrict__ agg,
                                const float* __restrict__ dn,
                                const float* __restrict__ bias,
                                float* __restrict__ out,
                                int lognf, int do_relu, int n_nodes) {
  long long i = (long long)blockIdx.x * blockDim.x + threadIdx.x;
  if (i >= ((long long)n_nodes << lognf)) return;
  int r = (int)(i >> lognf);
  int c = (int)(i & ((1 << lognf) - 1));
  float v = agg[i] * dn[r] + bias[c];
  out[i] = do_relu ? fmaxf(v, 0.0f) : v;
}

// ---------------- pooling GEMM: P[:,colOff:colOff+N] += L @ H --------------
// grid = (512/16 m-tiles, k-splits); atomic accumulate into pool buffer.
template <int N>
__global__ __launch_bounds__(32) void gemm_pool(
    const float* __restrict__ L,   // [512, NNODES]
    const float* __restrict__ H,   // [NNODES, N]
    float* __restrict__ P,         // [512, 96]
    int colOff, int kChunk) {
  const int lane = threadIdx.x;
  const int half = lane >> 4;
  const int lm   = lane & 15;
  const int m0   = blockIdx.x * 16;
  const int kBeg = blockIdx.y * kChunk;
  const int kEnd = kBeg + kChunk;      // kChunk divisible by 4
  constexpr int NT = N / 16;

  const float* Lrow = L + (size_t)(m0 + lm) * NNODES;

  v8f acc[NT] = {};
  for (int k0 = kBeg; k0 < kEnd; k0 += 4) {
    const int ka = k0 + 2 * half;
    v2f a;
    a.x = Lrow[ka + 0];
    a.y = Lrow[ka + 1];
#pragma unroll
    for (int t = 0; t < NT; ++t) {
      v2f b;
      b.x = H[(size_t)(ka + 0) * N + t * 16 + lm];
      b.y = H[(size_t)(ka + 1) * N + t * 16 + lm];
      acc[t] = wmma4(a, b, acc[t]);
    }
  }
#pragma unroll
  for (int t = 0; t < NT; ++t)
#pragma unroll
    for (int r = 0; r < 8; ++r)
      atomAddF(&P[(size_t)(m0 + r + 8 * half) * 96 + colOff + t * 16 + lm],
               acc[t][r]);
}

// ---------------- tiny MLP layers -----------------------------------------
__global__ void fc_kernel(const float* __restrict__ X, const float* __restrict__ W,
                          const float* __restrict__ b, float* __restrict__ Y,
                          int M, int K, int N, int do_relu) {
  int idx = blockIdx.x * blockDim.x + threadIdx.x;
  if (idx >= M * N) return;
  int r = idx / N, c = idx % N;
  float acc = b[c];
  for (int k = 0; k < K; ++k) acc += X[(size_t)r * K + k] * W[(size_t)k * N + c];
  Y[idx] = do_relu ? fmaxf(acc, 0.0f) : acc;
}

// ---------------------------------------------------------------------------
extern "C" void kernel_launch(void* const* d_in, const int* in_sizes, int n_in,
                              void* d_out, int out_size, void* d_ws, size_t ws_size,
                              hipStream_t stream) {
  const float* solute_x   = (const float*)d_in[0];
  const float* solvent_x  = (const float*)d_in[1];
  const float* solute_L   = (const float*)d_in[2];
  const float* solvent_L  = (const float*)d_in[3];
  const int*   solute_src = (const int*)d_in[4];
  const int*   solute_dst = (const int*)d_in[5];
  const int*   solvent_src= (const int*)d_in[6];
  const int*   solvent_dst= (const int*)d_in[7];
  const float* W1 = (const float*)d_in[8];
  const float* b1 = (const float*)d_in[9];
  const float* W2 = (const float*)d_in[10];
  const float* b2 = (const float*)d_in[11];
  const float* fc1_w = (const float*)d_in[12];
  const float* fc1_b = (const float*)d_in[13];
  const float* fc2_w = (const float*)d_in[14];
  const float* fc2_b = (const float*)d_in[15];
  const float* fc3_w = (const float*)d_in[16];
  const float* fc3_b = (const float*)d_in[17];

  const int NE = in_sizes[4];          // 3,200,000

  float* ws   = (float*)d_ws;
  float* sn   = ws + 0;                // [100000]
  float* dn   = ws + 100000;           // [100000]
  float* h    = ws + 200000;           // [100000,32] (reused for h2)
  float* agg  = ws + 3400000;          // [100000,32]
  float* su1  = ws + 6600000;          // [100000,32]
  float* su2  = ws + 9800000;          // [100000,16]
  float* pool = ws + 11400000;         // [512,96]
  float* m1   = ws + 11449152;         // [512,64]
  float* m2   = ws + 11481920;         // [512,16]

  hipMemsetAsync(pool, 0, (size_t)NGRAPH * 96 * sizeof(float), stream);

  for (int side = 0; side < 2; ++side) {
    const float* x   = side ? solvent_x   : solute_x;
    const float* L   = side ? solvent_L   : solute_L;
    const int*   src = side ? solvent_src : solute_src;
    const int*   dst = side ? solvent_dst : solute_dst;

    // degree norms (sn and dn are adjacent -> one memset)
    hipMemsetAsync(sn, 0, 2 * (size_t)NNODES * sizeof(float), stream);
    deg_kernel<<<(NE + 255) / 256, 256, 0, stream>>>(src, sn, NE);
    deg_kernel<<<(NE + 255) / 256, 256, 0, stream>>>(dst, dn, NE);
    rsqrt_kernel<<<(2 * NNODES + 255) / 256, 256, 0, stream>>>(sn, 2 * NNODES);

    // layer 1: h = (x ⊙ sn) @ W1   [100000,32]
    gemm_node_scaled<64, 32><<<NNODES / 16, 32, 0, stream>>>(x, sn, W1, h);
    hipMemsetAsync(agg, 0, (size_t)NNODES * 32 * sizeof(float), stream);
    scatter_kernel<<<(int)(((long long)NE * 32 + 255) / 256), 256, 0, stream>>>(
        h, src, dst, agg, NE, 5);
    epilogue_kernel<<<(NNODES * 32 + 255) / 256, 256, 0, stream>>>(
        agg, dn, b1, su1, 5, /*relu=*/1, NNODES);

    // layer 2: h2 = (su1 ⊙ sn) @ W2   [100000,16]
    gemm_node_scaled<32, 16><<<NNODES / 16, 32, 0, stream>>>(su1, sn, W2, h);
    hipMemsetAsync(agg, 0, (size_t)NNODES * 16 * sizeof(float), stream);
    scatter_kernel<<<(int)(((long long)NE * 16 + 255) / 256), 256, 0, stream>>>(
        h, src, dst, agg, NE, 4);
    epilogue_kernel<<<(NNODES * 16 + 255) / 256, 256, 0, stream>>>(
        agg, dn, b2, su2, 4, /*relu=*/0, NNODES);

    // pooling: pool[:, off] += L @ su{1,2}; K split 50 × 2000
    dim3 pg(NGRAPH / 16, 50);
    gemm_pool<32><<<pg, 32, 0, stream>>>(L, su1, pool, side ? 48 : 0, 2000);
    gemm_pool<16><<<pg, 32, 0, stream>>>(L, su2, pool, side ? 80 : 32, 2000);
  }

  // MLP head
  fc_kernel<<<(NGRAPH * 64 + 255) / 256, 256, 0, stream>>>(
      pool, fc1_w, fc1_b, m1, NGRAPH, 96, 64, 1);
  fc_kernel<<<(NGRAPH * 16 + 255) / 256, 256, 0, stream>>>(
      m1, fc2_w, fc2_b, m2, NGRAPH, 64, 16, 1);
  fc_kernel<<<(NGRAPH + 255) / 256, 256, 0, stream>>>(
      m2, fc3_w, fc3_b, (float*)d_out, NGRAPH, 16, 1, 0);
}